// GAT_49503793053817
// MI455X (gfx1250) — compile-verified
//
#include <hip/hip_runtime.h>
#include <hip/hip_bf16.h>
#include <math.h>

#define HID 64
#define NHEAD 4
#define CHAN 256            // NHEAD*HID
#define NEG_SLOPE 0.2f
#define LN_EPS 1e-5f

typedef float v2f __attribute__((ext_vector_type(2)));
typedef float v8f __attribute__((ext_vector_type(8)));

static __device__ __forceinline__ v8f wmma_f32(v2f a, v2f b, v8f c) {
  return __builtin_amdgcn_wmma_f32_16x16x4_f32(false, a, false, b, (short)0, c,
                                               false, false);
}

// ---------------------------------------------------------------------------
// Pre-pack W[K,M] row-major into WMMA B-fragment order so each lane's two
// K-values are adjacent (enables b64 loads):
//   Wp[((k>>2)*2 + ((k>>1)&1)) * 2M + col*2 + (k&1)] = W[k*M + col]
// ---------------------------------------------------------------------------
__global__ __launch_bounds__(256)
void k_prepack(const float* __restrict__ W, float* __restrict__ Wp,
               int K, int M) {
  const int i = blockIdx.x * blockDim.x + threadIdx.x;
  if (i >= K * M) return;
  const int k = i / M;
  const int col = i - k * M;
  const int dstIdx = ((k >> 2) * 2 + ((k >> 1) & 1)) * (2 * M) + col * 2 + (k & 1);
  Wp[dstIdx] = W[i];
}

// ---------------------------------------------------------------------------
// GEMM: Y[nrows,M] = X[nrows,K] @ W[K,M] (+ bias[M] if non-null), all f32,
// via V_WMMA_F32_16X16X4_F32. Each wave computes a 32x64 strip: two 16-row A
// fragments share four packed-B fragments -> 8 WMMAs per k-step, all loads
// base+immediate (K,M compile-time). Ragged last row-tile: clamped loads,
// guarded stores (EXEC all-ones through every WMMA).
// ---------------------------------------------------------------------------
template<int K, int M>
__global__ __launch_bounds__(128)
void k_gemm_wmma(const float* __restrict__ X, const float* __restrict__ Wp,
                 const float* __restrict__ bias, float* __restrict__ Y,
                 int nrows) {
  const int wave = (blockIdx.x * blockDim.x + threadIdx.x) >> 5;
  const int lane = threadIdx.x & 31;
  constexpr int nColGroups = M >> 6;
  const int nRowTiles = (nrows + 31) >> 5;
  const int tileM = wave / nColGroups;
  const int cg    = wave - tileM * nColGroups;
  if (tileM >= nRowTiles) return;            // wave-uniform exit

  const int row0 = tileM * 32;
  const int col0 = cg * 64;
  const int lmod = lane & 15;
  const int ldiv = lane >> 4;                // 0: K pair {0,1}, 1: K pair {2,3}

  const int rA0 = min(row0 + lmod,      nrows - 1);
  const int rA1 = min(row0 + 16 + lmod, nrows - 1);
  const float* x0 = X + (size_t)rA0 * K + 2 * ldiv;
  const float* x1 = X + (size_t)rA1 * K + 2 * ldiv;
  const float* wb = Wp + ((size_t)ldiv * M + col0 + lmod) * 2;

  v8f c0 = {}, c1 = {}, c2 = {}, c3 = {};    // rows row0..row0+15
  v8f c4 = {}, c5 = {}, c6 = {}, c7 = {};    // rows row0+16..row0+31

  #pragma unroll
  for (int k0 = 0; k0 < K; k0 += 4) {
    const v2f a0 = *(const v2f*)(x0 + k0);
    const v2f a1 = *(const v2f*)(x1 + k0);
    const float* wk = wb + (size_t)(k0 >> 2) * (4 * M);
    const v2f b0 = *(const v2f*)(wk);
    const v2f b1 = *(const v2f*)(wk + 32);
    const v2f b2 = *(const v2f*)(wk + 64);
    const v2f b3 = *(const v2f*)(wk + 96);
    c0 = wmma_f32(a0, b0, c0);
    c1 = wmma_f32(a0, b1, c1);
    c2 = wmma_f32(a0, b2, c2);
    c3 = wmma_f32(a0, b3, c3);
    c4 = wmma_f32(a1, b0, c4);
    c5 = wmma_f32(a1, b1, c5);
    c6 = wmma_f32(a1, b2, c6);
    c7 = wmma_f32(a1, b3, c7);
  }

  const float bb0 = bias ? bias[col0 + lmod]      : 0.0f;
  const float bb1 = bias ? bias[col0 + 16 + lmod] : 0.0f;
  const float bb2 = bias ? bias[col0 + 32 + lmod] : 0.0f;
  const float bb3 = bias ? bias[col0 + 48 + lmod] : 0.0f;

  #pragma unroll
  for (int v = 0; v < 8; ++v) {
    const int r0 = row0 + v + 8 * ldiv;       // frag0: VGPR v -> rows v, v+8
    const int r1 = r0 + 16;                   // frag1
    if (r0 < nrows) {
      float* yr = Y + (size_t)r0 * M + col0 + lmod;
      yr[0]  = c0[v] + bb0;
      yr[16] = c1[v] + bb1;
      yr[32] = c2[v] + bb2;
      yr[48] = c3[v] + bb3;
    }
    if (r1 < nrows) {
      float* yr = Y + (size_t)r1 * M + col0 + lmod;
      yr[0]  = c4[v] + bb0;
      yr[16] = c5[v] + bb1;
      yr[32] = c6[v] + bb2;
      yr[48] = c7[v] + bb3;
    }
  }
}

// ---------------------------------------------------------------------------
// el[n,h] = sum_d fs[n,h,d]*al[h,d];  er likewise with ar. Wave per node.
// ---------------------------------------------------------------------------
__global__ __launch_bounds__(256)
void k_el_er(const float* __restrict__ fs, const float* __restrict__ al,
             const float* __restrict__ ar, float* __restrict__ el,
             float* __restrict__ er, int n_nodes) {
  const int wave = (blockIdx.x * blockDim.x + threadIdx.x) >> 5;
  const int lane = threadIdx.x & 31;
  if (wave >= n_nodes) return;
  const int c0 = lane * 8;                   // channel base, head = lane>>3
  const float* row = fs + (size_t)wave * CHAN + c0;
  float sl = 0.0f, sr = 0.0f;
  #pragma unroll
  for (int k = 0; k < 8; ++k) {
    const float v = row[k];
    sl += v * al[c0 + k];
    sr += v * ar[c0 + k];
  }
  #pragma unroll
  for (int m = 1; m < 8; m <<= 1) {
    sl += __shfl_xor(sl, m, 32);
    sr += __shfl_xor(sr, m, 32);
  }
  if ((lane & 7) == 0) {
    const int h = lane >> 3;
    el[wave * NHEAD + h] = sl;
    er[wave * NHEAD + h] = sr;
  }
}

// CSR row pointers over sorted dst via binary search (thread per node).
__global__ __launch_bounds__(256)
void k_rowptr(const int* __restrict__ dst, int* __restrict__ rowptr,
              int n_nodes, int n_edges) {
  const int n = blockIdx.x * blockDim.x + threadIdx.x;
  if (n > n_nodes) return;
  int lo = 0, hi = n_edges;
  while (lo < hi) {
    const int mid = (lo + hi) >> 1;
    if (dst[mid] < n) lo = mid + 1; else hi = mid;
  }
  rowptr[n] = lo;
}

// e[e,h] = leaky_relu(el[src[e],h] + er[dst[e],h]); thread per edge (float4).
__global__ __launch_bounds__(256)
void k_edge_scores(const float* __restrict__ el, const float* __restrict__ er,
                   const int* __restrict__ src, const int* __restrict__ dst,
                   float* __restrict__ ebuf, int n_edges) {
  const int e = blockIdx.x * blockDim.x + threadIdx.x;
  if (e >= n_edges) return;
  const float4 l = *(const float4*)(el + (size_t)src[e] * NHEAD);
  const float4 r = *(const float4*)(er + (size_t)dst[e] * NHEAD);
  float4 o;
  o.x = l.x + r.x; o.y = l.y + r.y; o.z = l.z + r.z; o.w = l.w + r.w;
  o.x = o.x >= 0.0f ? o.x : NEG_SLOPE * o.x;
  o.y = o.y >= 0.0f ? o.y : NEG_SLOPE * o.y;
  o.z = o.z >= 0.0f ? o.z : NEG_SLOPE * o.z;
  o.w = o.w >= 0.0f ? o.w : NEG_SLOPE * o.w;
  *(float4*)(ebuf + (size_t)e * NHEAD) = o;
}

// In-place edge softmax over each node's incoming edges. Wave per node:
// lane -> (head = lane&3, edge slot stride 8). Head groups reduced via xor 4,8,16.
__global__ __launch_bounds__(256)
void k_softmax(float* __restrict__ ebuf, const int* __restrict__ rowptr,
               int n_nodes) {
  const int wave = (blockIdx.x * blockDim.x + threadIdx.x) >> 5;
  const int lane = threadIdx.x & 31;
  if (wave >= n_nodes) return;
  const int start = rowptr[wave];
  const int deg   = rowptr[wave + 1] - start;
  if (deg == 0) return;
  const int h  = lane & 3;
  const int j0 = lane >> 2;

  float m = -INFINITY;
  for (int j = j0; j < deg; j += 8)
    m = fmaxf(m, ebuf[(size_t)(start + j) * NHEAD + h]);
  #pragma unroll
  for (int mask = 4; mask < 32; mask <<= 1) m = fmaxf(m, __shfl_xor(m, mask, 32));

  float s = 0.0f;
  for (int j = j0; j < deg; j += 8)
    s += __expf(ebuf[(size_t)(start + j) * NHEAD + h] - m);
  #pragma unroll
  for (int mask = 4; mask < 32; mask <<= 1) s += __shfl_xor(s, mask, 32);

  const float inv = 1.0f / s;
  for (int j = j0; j < deg; j += 8) {
    const size_t idx = (size_t)(start + j) * NHEAD + h;
    ebuf[idx] = __expf(ebuf[idx] - m) * inv;
  }
}

// agg[n,c] = sum_{e in edges(n)} a[e,head(c)] * fs[src[e],c]. Wave per node,
// lane owns 8 contiguous channels (wave reads 1KB contiguous per edge).
__global__ __launch_bounds__(256)
void k_aggregate(const float* __restrict__ abuf, const float* __restrict__ fs,
                 const int* __restrict__ src, const int* __restrict__ rowptr,
                 float* __restrict__ agg, int n_nodes) {
  const int wave = (blockIdx.x * blockDim.x + threadIdx.x) >> 5;
  const int lane = threadIdx.x & 31;
  if (wave >= n_nodes) return;
  const int start = rowptr[wave];
  const int deg   = rowptr[wave + 1] - start;
  const int c0    = lane * 8;
  const int head  = lane >> 3;

  float acc[8] = {0,0,0,0,0,0,0,0};
  for (int j = 0; j < deg; ++j) {
    const int e   = start + j;
    const float a = abuf[(size_t)e * NHEAD + head];
    const float* fr = fs + (size_t)src[e] * CHAN + c0;
    const float4 f0 = *(const float4*)fr;
    const float4 f1 = *(const float4*)(fr + 4);
    acc[0] += a * f0.x; acc[1] += a * f0.y; acc[2] += a * f0.z; acc[3] += a * f0.w;
    acc[4] += a * f1.x; acc[5] += a * f1.y; acc[6] += a * f1.z; acc[7] += a * f1.w;
  }
  float* out = agg + (size_t)wave * CHAN + c0;
  *(float4*)out       = make_float4(acc[0], acc[1], acc[2], acc[3]);
  *(float4*)(out + 4) = make_float4(acc[4], acc[5], acc[6], acc[7]);
}

// h_out = relu(LN(mean_heads(agg) + resid) * g + b). Wave per node, 2 ch/lane.
__global__ __launch_bounds__(256)
void k_post(const float* __restrict__ agg, const float* __restrict__ resid,
            const float* __restrict__ g, const float* __restrict__ b,
            float* __restrict__ h_out, int n_nodes) {
  const int wave = (blockIdx.x * blockDim.x + threadIdx.x) >> 5;
  const int lane = threadIdx.x & 31;
  if (wave >= n_nodes) return;
  const float* ar = agg + (size_t)wave * CHAN;
  const float* rr = resid + (size_t)wave * HID;
  float v[2];
  #pragma unroll
  for (int t = 0; t < 2; ++t) {
    const int d = lane * 2 + t;
    v[t] = 0.25f * (ar[d] + ar[64 + d] + ar[128 + d] + ar[192 + d]) + rr[d];
  }
  float s = v[0] + v[1];
  #pragma unroll
  for (int m = 1; m < 32; m <<= 1) s += __shfl_xor(s, m, 32);
  const float mu = s * (1.0f / 64.0f);
  const float d0 = v[0] - mu, d1 = v[1] - mu;
  float q = d0 * d0 + d1 * d1;
  #pragma unroll
  for (int m = 1; m < 32; m <<= 1) q += __shfl_xor(q, m, 32);
  const float rs = rsqrtf(q * (1.0f / 64.0f) + LN_EPS);
  float* outr = h_out + (size_t)wave * HID;
  {
    const int da = lane * 2;
    const float oa = d0 * rs * g[da] + b[da];
    const float ob = d1 * rs * g[da + 1] + b[da + 1];
    outr[da]     = oa > 0.0f ? oa : 0.0f;
    outr[da + 1] = ob > 0.0f ? ob : 0.0f;
  }
}

// ---------------------------------------------------------------------------
template<int K, int M>
static void launch_gemm(const float* X, const float* W, const float* bias,
                        float* Y, float* Wp, int N, hipStream_t stream) {
  k_prepack<<<(K * M + 255) / 256, 256, 0, stream>>>(W, Wp, K, M);
  const int waves = ((N + 31) / 32) * (M / 64);
  k_gemm_wmma<K, M><<<(waves + 3) / 4, 128, 0, stream>>>(X, Wp, bias, Y, N);
}

extern "C" void kernel_launch(void* const* d_in, const int* in_sizes, int n_in,
                              void* d_out, int out_size, void* d_ws, size_t ws_size,
                              hipStream_t stream) {
  const float* feats = (const float*)d_in[0];
  const int*   src   = (const int*)  d_in[1];
  const int*   dst   = (const int*)  d_in[2];
  const float* W0    = (const float*)d_in[3];
  const float* al0   = (const float*)d_in[4];
  const float* ar0   = (const float*)d_in[5];
  const float* resw0 = (const float*)d_in[6];
  const float* resb0 = (const float*)d_in[7];
  const float* g0    = (const float*)d_in[8];
  const float* b0    = (const float*)d_in[9];
  const float* W1    = (const float*)d_in[10];
  const float* al1   = (const float*)d_in[11];
  const float* ar1   = (const float*)d_in[12];
  const float* resw1 = (const float*)d_in[13];
  const float* resb1 = (const float*)d_in[14];
  const float* g1    = (const float*)d_in[15];
  const float* b1    = (const float*)d_in[16];
  const float* predw = (const float*)d_in[17];
  const float* predb = (const float*)d_in[18];
  float* out = (float*)d_out;

  const int IN = 128;
  const int N  = in_sizes[0] / IN;    // 50000
  const int E  = in_sizes[1];         // 800000

  // workspace carve-up (all f32 unless noted; every offset is even -> 8B align)
  float* ws    = (float*)d_ws;
  float* fs    = ws;                                   // N*256
  float* agg   = fs    + (size_t)N * CHAN;             // N*256
  float* ebuf  = agg   + (size_t)N * CHAN;             // E*4
  float* el    = ebuf  + (size_t)E * NHEAD;            // N*4
  float* er    = el    + (size_t)N * NHEAD;            // N*4
  float* resid = er    + (size_t)N * NHEAD;            // N*64
  float* h0    = resid + (size_t)N * HID;              // N*64
  float* h1    = h0    + (size_t)N * HID;              // N*64
  float* Wp    = h1    + (size_t)N * HID;              // 128*256 (max packed W)
  int*   rowptr = (int*)(Wp + 128 * 256);              // N+1

  const int nodeBlocks = (N + 7) / 8;                  // wave-per-node kernels

  // graph CSR (shared by both layers)
  k_rowptr<<<(N + 256) / 256, 256, 0, stream>>>(dst, rowptr, N, E);

  // ---------------- layer 0 ----------------
  launch_gemm<128, 256>(feats, W0, nullptr, fs, Wp, N, stream);
  launch_gemm<128, 64>(feats, resw0, resb0, resid, Wp, N, stream);
  k_el_er<<<nodeBlocks, 256, 0, stream>>>(fs, al0, ar0, el, er, N);
  k_edge_scores<<<(E + 255) / 256, 256, 0, stream>>>(el, er, src, dst, ebuf, E);
  k_softmax<<<nodeBlocks, 256, 0, stream>>>(ebuf, rowptr, N);
  k_aggregate<<<nodeBlocks, 256, 0, stream>>>(ebuf, fs, src, rowptr, agg, N);
  k_post<<<nodeBlocks, 256, 0, stream>>>(agg, resid, g0, b0, h0, N);

  // ---------------- layer 1 ----------------
  launch_gemm<64, 256>(h0, W1, nullptr, fs, Wp, N, stream);
  launch_gemm<64, 64>(h0, resw1, resb1, resid, Wp, N, stream);
  k_el_er<<<nodeBlocks, 256, 0, stream>>>(fs, al1, ar1, el, er, N);
  k_edge_scores<<<(E + 255) / 256, 256, 0, stream>>>(el, er, src, dst, ebuf, E);
  k_softmax<<<nodeBlocks, 256, 0, stream>>>(ebuf, rowptr, N);
  k_aggregate<<<nodeBlocks, 256, 0, stream>>>(ebuf, fs, src, rowptr, agg, N);
  k_post<<<nodeBlocks, 256, 0, stream>>>(agg, resid, g1, b1, h1, N);

  // ---------------- prediction head ----------------
  launch_gemm<64, 64>(h1, predw, predb, out, Wp, N, stream);
}